// VectorQuantizer_54906861912274
// MI455X (gfx1250) — compile-verified
//
#include <hip/hip_runtime.h>
#include <hip/hip_bf16.h>
#include <math.h>

// ---------------------------------------------------------------------------
// VQ-VAE vector quantizer for MI455X (gfx1250, wave32, WMMA).
//
// Sizes: B=16, D=64, H=W=64 -> N = 65536 rows, K = 1024 codes.
// Memory layout quirk: "flat" x[n,d] = inputs[b, d, hw]  (n = b*4096 + hw),
// i.e. rows of the GEMM are columns of the contiguous [B,D,HW] tensor; 16
// consecutive n's are 16 contiguous floats for each d.
//
// d_out layout (concatenated, float32):
//   [0]                 loss                      (1)
//   [1        .. ]      quantized (NCHW)          (4194304)
//   [4194305]           perplexity                (1)
//   [4194306  .. ]      encodings one-hot [N,K]   (67108864)
//   [71303170 .. ]      indices (as float)        (65536)
// ---------------------------------------------------------------------------

typedef float v2f __attribute__((ext_vector_type(2)));
typedef float v8f __attribute__((ext_vector_type(8)));

#define N_ROWS   65536
#define EMB_K    1024
#define EMB_D    64
#define HW       4096
#define Q_ELEMS  4194304   // B*D*H*W

// ---- codebook squared norms: enorm[k] = sum_d emb[k,d]^2 -------------------
__global__ void vq_enorm_kernel(const float* __restrict__ emb,
                                float* __restrict__ enorm) {
    int k = blockIdx.x * blockDim.x + threadIdx.x;
    if (k < EMB_K) {
        const float* r = emb + ((size_t)k << 6);
        float s = 0.f;
#pragma unroll
        for (int d = 0; d < EMB_D; ++d) s += r[d] * r[d];
        enorm[k] = s;
    }
}

// ---- WMMA distance GEMM + argmin ------------------------------------------
// One wave per 16-row tile; 8 waves (256 threads) per block; 512 blocks.
// score[n,k] = -2 * x[n]·e[k] + ||e[k]||^2  (||x||^2 dropped: constant per n)
__global__ __launch_bounds__(256) void vq_wmma_argmin_kernel(
    const float* __restrict__ in,      // [B, D, HW]
    const float* __restrict__ emb,     // [K, D]
    const float* __restrict__ enorm,   // [K]
    float* __restrict__ idx_out)       // [N] (float-coded indices)
{
    const int tid  = threadIdx.x;
    const int wave = tid >> 5;
    const int lane = tid & 31;
    const int m    = lane & 15;   // row within tile (A) / column (B,C)
    const int kg   = lane >> 4;   // K-group half for A/B, row-half for C

    const int tile = blockIdx.x * 8 + wave;   // 0..4095
    const int n0   = tile << 4;               // first row of tile
    const int b    = n0 >> 12;                // n0 / HW
    const int hw0  = n0 & (HW - 1);

    // Preload A tile (16 rows x 64 dims) scaled by -2, in WMMA f32 16x16x4
    // A layout: lane = kg*16 + m holds (row m, K = 2*kg + {0,1}) per step.
    const float* abase = in + ((size_t)b << 18) + hw0 + m;   // b*D*HW
    v2f a[16];
#pragma unroll
    for (int t = 0; t < 16; ++t) {
        const int d = 4 * t + 2 * kg;
        a[t].x = -2.0f * abase[(size_t)d << 12];
        a[t].y = -2.0f * abase[(size_t)(d + 1) << 12];
    }

    float best[8];
    int   bidx[8];
#pragma unroll
    for (int j = 0; j < 8; ++j) { best[j] = 3.4e38f; bidx[j] = 0; }

    // Sweep 64 column tiles of 16 codebook entries each.
    for (int c = 0; c < 64; ++c) {
        const int    kcol = (c << 4) + m;                       // this lane's column
        const float* bp   = emb + ((size_t)kcol << 6) + 2 * kg; // row kcol, dim base

        v8f acc = {0.f, 0.f, 0.f, 0.f, 0.f, 0.f, 0.f, 0.f};
#pragma unroll
        for (int t = 0; t < 16; ++t) {
            // B layout: lane kg*16+m holds (K = 2*kg + {0,1}, col m) -> float2 load
            v2f bv = *(const v2f*)(bp + 4 * t);
            acc = __builtin_amdgcn_wmma_f32_16x16x4_f32(
                false, a[t], false, bv, (short)0, acc, false, false);
        }

        const float en = enorm[kcol];
        // C layout: VGPR j holds row (j + 8*kg), column = lane&15
#pragma unroll
        for (int j = 0; j < 8; ++j) {
            const float s = acc[j] + en;
            if (s < best[j]) { best[j] = s; bidx[j] = kcol; }
        }
    }

    // Min-reduce across the 16 lanes that share the same 8 rows.
    // Tie-break on lowest index to match jnp.argmin.
#pragma unroll
    for (int j = 0; j < 8; ++j) {
#pragma unroll
        for (int mask = 1; mask < 16; mask <<= 1) {
            const float ov = __shfl_xor(best[j], mask, 32);
            const int   oi = __shfl_xor(bidx[j], mask, 32);
            if (ov < best[j] || (ov == best[j] && oi < bidx[j])) {
                best[j] = ov; bidx[j] = oi;
            }
        }
    }

    if (m == 0) {
#pragma unroll
        for (int j = 0; j < 8; ++j)
            idx_out[n0 + j + 8 * kg] = (float)bidx[j];
    }
}

// ---- gather quantized (NCHW) + loss partial sums --------------------------
__global__ __launch_bounds__(256) void vq_quantize_kernel(
    const float* __restrict__ in,     // [B, D, HW]
    const float* __restrict__ emb,    // [K, D]
    const float* __restrict__ idx_f,  // [N]
    float* __restrict__ qout,         // [B, D, HW]
    float* __restrict__ loss_acc)
{
    __shared__ float red[256];
    const int g   = blockIdx.x * 256 + threadIdx.x;   // over B*D*HW
    const int b   = g >> 18;
    const int rem = g & 262143;
    const int d   = rem >> 12;
    const int hw  = rem & (HW - 1);
    const int n   = (b << 12) + hw;

    const int   k = (int)idx_f[n];
    const float q = emb[((size_t)k << 6) + d];
    const float x = in[g];
    qout[g] = q;                       // straight-through value == quantized
    const float df = q - x;
    red[threadIdx.x] = df * df;
    __syncthreads();
    for (int s = 128; s > 0; s >>= 1) {
        if (threadIdx.x < s) red[threadIdx.x] += red[threadIdx.x + s];
        __syncthreads();
    }
    if (threadIdx.x == 0) atomicAdd(loss_acc, red[0]);
}

// ---- one-hot scatter + histogram ------------------------------------------
__global__ void vq_onehot_kernel(const float* __restrict__ idx_f,
                                 float* __restrict__ enc,
                                 unsigned int* __restrict__ counts) {
    const int n = blockIdx.x * blockDim.x + threadIdx.x;
    if (n < N_ROWS) {
        const int k = (int)idx_f[n];
        enc[((size_t)n << 10) + k] = 1.0f;
        atomicAdd(&counts[k], 1u);
    }
}

// ---- finalize: loss scalar + perplexity -----------------------------------
__global__ __launch_bounds__(256) void vq_finalize_kernel(
    const unsigned int* __restrict__ counts,
    const float* __restrict__ loss_acc,
    float* __restrict__ loss_out,
    float* __restrict__ perp_out)
{
    __shared__ float red[256];
    float h = 0.f;
    for (int i = threadIdx.x; i < EMB_K; i += 256) {
        const float p = (float)counts[i] * (1.0f / (float)N_ROWS);
        h += p * logf(p + 1e-10f);
    }
    red[threadIdx.x] = h;
    __syncthreads();
    for (int s = 128; s > 0; s >>= 1) {
        if (threadIdx.x < s) red[threadIdx.x] += red[threadIdx.x + s];
        __syncthreads();
    }
    if (threadIdx.x == 0) {
        *perp_out = expf(-red[0]);
        // loss = q_latent + 0.25 * e_latent, both equal mean((q-x)^2)
        *loss_out = 1.25f * loss_acc[0] / (float)Q_ELEMS;
    }
}

extern "C" void kernel_launch(void* const* d_in, const int* in_sizes, int n_in,
                              void* d_out, int out_size, void* d_ws, size_t ws_size,
                              hipStream_t stream) {
    (void)in_sizes; (void)n_in; (void)out_size; (void)ws_size;
    const float* in  = (const float*)d_in[0];   // [16,64,64,64] f32
    const float* emb = (const float*)d_in[1];   // [1024,64] f32

    float* out      = (float*)d_out;
    float* loss_out = out;                       // [1]
    float* qout     = out + 1;                   // [4194304]
    float* perp_out = out + 4194305;             // [1]
    float* enc      = out + 4194306;             // [67108864]
    float* idx_f    = out + 71303170;            // [65536] indices (float)

    float*        ws_enorm = (float*)d_ws;                        // 1024 f32
    unsigned int* counts   = (unsigned int*)((char*)d_ws + 4096); // 1024 u32
    float*        loss_acc = (float*)((char*)d_ws + 8192);        // 1 f32

    // zero the one-hot region and the accumulators (stream-ordered, capturable)
    hipMemsetAsync(enc, 0, (size_t)67108864 * sizeof(float), stream);
    hipMemsetAsync((char*)d_ws + 4096, 0, 4096 + 16, stream);

    vq_enorm_kernel<<<4, 256, 0, stream>>>(emb, ws_enorm);
    vq_wmma_argmin_kernel<<<512, 256, 0, stream>>>(in, emb, ws_enorm, idx_f);
    vq_quantize_kernel<<<Q_ELEMS / 256, 256, 0, stream>>>(in, emb, idx_f, qout, loss_acc);
    vq_onehot_kernel<<<N_ROWS / 256, 256, 0, stream>>>(idx_f, enc, counts);
    vq_finalize_kernel<<<1, 256, 0, stream>>>(counts, loss_acc, loss_out, perp_out);
}